// PrimitiveEquationsBlockStepper_33165737460147
// MI455X (gfx1250) — compile-verified
//
#include <hip/hip_runtime.h>
#include <hip/hip_bf16.h>

typedef __attribute__((ext_vector_type(2))) float v2f;
typedef __attribute__((ext_vector_type(8))) float v8f;

#define BB 2
#define KL 3
#define HH 256
#define WW 512
#define NSX 19            // scalar-field channels (x_s)
#define NOUT 25           // 19 scalar + 6 vector outputs
#define NCH 26            // 19 + 6 + 1 zero pad (even for pair loads)
#define NPAIR (NCH / 2)
#define TILEH 8
#define TILEW 16
#define HALOH (TILEH + 4)
#define HALOW (TILEW + 4)
#define RGASC 287.0f
#define CPC   1004.0f

static const size_t PLANE = (size_t)HH * WW;           // 131072
static const size_t OFFT  = (size_t)BB * KL * PLANE * 2; // uv elements in d_out
static const size_t OFFQ  = OFFT + (size_t)BB * KL * PLANE;

// ---------------------------------------------------------------------------
// Combined dense weight: Wc[o(32 pad)][i(28 pad)][tap], pw_ss folded into the
// center tap of the k_ss block.  Stored pre-swizzled into the WMMA A-matrix
// lane layout: lane = m + 16*(k>>1), vgpr = k&1  (16x4 f32 A).
// ---------------------------------------------------------------------------
__device__ __forceinline__ float wc_elem(int o, int i, int tap,
                                         const float* kss, const float* kvs,
                                         const float* ksv, const float* kvv,
                                         const float* pw) {
    if (o < NSX && i < NSX) {
        float w = kss[(o * NSX + i) * 25 + tap];
        if (tap == 12) w += pw[o * NSX + i];
        return w;
    }
    if (o < NSX && i >= NSX && i < NOUT)  return kvs[(o * 6 + (i - NSX)) * 25 + tap];
    if (o >= NSX && o < NOUT && i < NSX)  return ksv[((o - NSX) * NSX + i) * 25 + tap];
    if (o >= NSX && o < NOUT && i >= NSX && i < NOUT)
        return kvv[((o - NSX) * 6 + (i - NSX)) * 25 + tap];
    return 0.0f;
}

__global__ void wprep_kernel(const float* __restrict__ kss, const float* __restrict__ kvs,
                             const float* __restrict__ ksv, const float* __restrict__ kvv,
                             const float* __restrict__ pw, float* __restrict__ WA) {
    int e = blockIdx.x * 256 + threadIdx.x;      // entry = [tap][ks][mb][lane]
    if (e >= 25 * 7 * 2 * 32) return;
    int lane = e & 31;
    int mb   = (e >> 5) & 1;
    int ks   = (e >> 6) % 7;
    int tap  = e / 448;
    int o  = mb * 16 + (lane & 15);
    int i0 = ks * 4 + 2 * (lane >> 4);
    WA[2 * e + 0] = wc_elem(o, i0 + 0, tap, kss, kvs, ksv, kvv, pw);
    WA[2 * e + 1] = wc_elem(o, i0 + 1, tap, kss, kvs, ksv, kvv, pw);
}

// ---------------------------------------------------------------------------
// Per (b,level) spatial mean of T.
// ---------------------------------------------------------------------------
__global__ void mean_kernel(const float* __restrict__ Tsrc, float* __restrict__ Tmean) {
    __shared__ float red[256];
    const float* base = Tsrc + (size_t)blockIdx.x * PLANE;
    float s = 0.0f;
    for (int i = threadIdx.x; i < (int)PLANE; i += 256) s += base[i];
    red[threadIdx.x] = s;
    __syncthreads();
    for (int st = 128; st > 0; st >>= 1) {
        if ((int)threadIdx.x < st) red[threadIdx.x] += red[threadIdx.x + st];
        __syncthreads();
    }
    if (threadIdx.x == 0) Tmean[blockIdx.x] = red[0] * (1.0f / (float)PLANE);
}

// ---------------------------------------------------------------------------
// Build the 26-channel field X from the current RK-stage state.
//   ch l*6+0 : T - Tmean      ch l*6+1 : ke       ch +2,+3 : 0
//   ch l*6+4 : T              ch l*6+5 : q
//   ch 18    : f_cor(y)       ch 19+l*2+c : uv    ch 25 : 0
// ---------------------------------------------------------------------------
__global__ void prep_kernel(const float* __restrict__ uvS, const float* __restrict__ Tsrc,
                            const float* __restrict__ qSrc, const float* __restrict__ Tmean,
                            const float* __restrict__ kew, const float* __restrict__ fcor,
                            float* __restrict__ Xg) {
    int idx = blockIdx.x * 256 + threadIdx.x;
    if (idx >= BB * (int)PLANE) return;
    int b   = idx / (int)PLANE;
    int rem = idx % (int)PLANE;
    int y = rem / WW, x = rem % WW;

    float u[KL], v[KL], Tt[KL], qq[KL];
    for (int l = 0; l < KL; ++l) {
        size_t base = ((size_t)(b * KL + l) * HH + y) * WW + x;
        u[l]  = uvS[base * 2 + 0];
        v[l]  = uvS[base * 2 + 1];
        Tt[l] = Tsrc[base];
        qq[l] = qSrc[base];
    }
    size_t pix = (size_t)y * WW + x;
    float* Xb = Xg + (size_t)b * NCH * PLANE + pix;
#define XW(ch, val) Xb[(size_t)(ch) * PLANE] = (val)
    for (int l = 0; l < KL; ++l) {
        float ke = 0.0f;
        for (int i = 0; i < KL; ++i)
            ke += kew[l * KL + i] * (u[i] * u[i] + v[i] * v[i]);
        XW(l * 6 + 0, Tt[l] - Tmean[b * KL + l]);
        XW(l * 6 + 1, ke);
        XW(l * 6 + 2, 0.0f);
        XW(l * 6 + 3, 0.0f);
        XW(l * 6 + 4, Tt[l]);
        XW(l * 6 + 5, qq[l]);
        XW(NSX + l * 2 + 0 + 0, u[l]);   // ch 19+2l
        XW(NSX + l * 2 + 1 + 0, v[l]);   // ch 20+2l
    }
    XW(18, fcor[y]);
    XW(25, 0.0f);
#undef XW
}

// ---------------------------------------------------------------------------
// Main kernel: implicit-GEMM 5x5 conv via V_WMMA_F32_16X16X4_F32, plus grouped
// conv (k_vs2), svp/Coriolis, vertical physics (omega, dT/dp) and RK4 fusion.
// One 256-thread workgroup handles a 16(W) x 8(H) pixel tile of one batch.
// ---------------------------------------------------------------------------
__global__ __launch_bounds__(256)
void conv_kernel(const float* __restrict__ Xg, const float* __restrict__ WA,
                 const float* __restrict__ kvs2, const float* __restrict__ svp,
                 const float* __restrict__ plev, const float* __restrict__ dpv,
                 const int* __restrict__ dtp,
                 const float* __restrict__ oUV, const float* __restrict__ oT,
                 const float* __restrict__ oQ,
                 float* __restrict__ aUV, float* __restrict__ aT, float* __restrict__ aQ,
                 float* __restrict__ nUV, float* __restrict__ nT, float* __restrict__ nQ,
                 float* __restrict__ out, int stage, float wgt, float aCoef) {
    __shared__ float Xs[NPAIR * HALOH * HALOW * 2];   // pair-interleaved halo tile
    __shared__ float Ys[NOUT * TILEH * TILEW];        // conv results

    const int t   = threadIdx.x;
    const int b   = blockIdx.z;
    const int ty0 = blockIdx.y * TILEH;
    const int tx0 = blockIdx.x * TILEW;

    // ---- stage halo tile into LDS: zero-pad H, wrap W --------------------
    for (int s = t; s < NPAIR * HALOH * HALOW; s += 256) {
        int p   = s / (HALOH * HALOW);
        int rem = s % (HALOH * HALOW);
        int py = rem / HALOW, px = rem % HALOW;
        int gy = ty0 + py - 2;
        int gx = (tx0 + px - 2 + WW) % WW;
        float v0 = 0.0f, v1 = 0.0f;
        if (gy >= 0 && gy < HH) {
            const float* base = Xg + (((size_t)b * NCH + 2 * p) * HH + gy) * WW + gx;
            v0 = base[0];
            v1 = base[PLANE];
        }
        Xs[2 * s + 0] = v0;
        Xs[2 * s + 1] = v1;
    }
    __syncthreads();

    // ---- WMMA implicit GEMM: M=out-ch(2x16), N=16 pixels, K=28 in 4-steps -
    const int lane = t & 31;
    const int g    = t >> 5;          // wave -> output row 0..7
    const int n15  = lane & 15;       // pixel column within row
    const int lhi  = lane >> 4;       // K lane-group
    v8f acc0 = {0.f, 0.f, 0.f, 0.f, 0.f, 0.f, 0.f, 0.f};
    v8f acc1 = {0.f, 0.f, 0.f, 0.f, 0.f, 0.f, 0.f, 0.f};
#pragma unroll 1
    for (int tap = 0; tap < 25; ++tap) {
        const int dy = tap / 5, dx = tap % 5;
        const int py = g + dy;
        const int px = n15 + dx;
#pragma unroll
        for (int ks = 0; ks < 7; ++ks) {
            const int pair = ks * 2 + lhi;                        // inch pair = (4ks+2*lhi)/2
            v2f bm = *(const v2f*)&Xs[((pair * HALOH + py) * HALOW + px) * 2];
            const float* ap = WA + ((size_t)((tap * 7 + ks) * 2) * 32 + lane) * 2;
            v2f a0 = *(const v2f*)(ap);
            v2f a1 = *(const v2f*)(ap + 64);                      // mb=1 block
            acc0 = __builtin_amdgcn_wmma_f32_16x16x4_f32(false, a0, false, bm,
                                                         (short)0, acc0, false, false);
            acc1 = __builtin_amdgcn_wmma_f32_16x16x4_f32(false, a1, false, bm,
                                                         (short)0, acc1, false, false);
        }
    }
    // D layout: comp r, lane L -> m = r + 8*(L>>4), n = L&15
#pragma unroll
    for (int r = 0; r < 8; ++r) {
        int oc0 = r + 8 * lhi;
        Ys[(oc0 * TILEH + g) * TILEW + n15] = acc0[r];
        int oc1 = 16 + r + 8 * lhi;
        if (oc1 < NOUT) Ys[(oc1 * TILEH + g) * TILEW + n15] = acc1[r];
    }
    __syncthreads();

    // ---- per-pixel physics (128 pixels -> threads 0..127) ----------------
    if (t < TILEH * TILEW) {
        const int px = t & 15, py = t >> 4;
        const float dtf = (float)(*dtp);
#define XR(ch, hy, hx) Xs[((((ch) >> 1) * HALOH + (hy)) * HALOW + (hx)) * 2 + ((ch) & 1)]
        float yo[NOUT];
#pragma unroll
        for (int o = 0; o < NOUT; ++o) yo[o] = Ys[(o * TILEH + py) * TILEW + px];

        // grouped conv (prod term): y_s[o] += sum_{c,tap} w2[o,c,tap]*xv[c]*x_s[o]
#pragma unroll 1
        for (int tap = 0; tap < 25; ++tap) {
            const int hy = py + tap / 5, hx = px + tap % 5;
            float xv6[6];
#pragma unroll
            for (int c = 0; c < 6; ++c) xv6[c] = XR(NSX + c, hy, hx);
#pragma unroll 1
            for (int o = 0; o < NSX; ++o) {
                float w = 0.0f;
#pragma unroll
                for (int c = 0; c < 6; ++c) w += kvs2[(o * 6 + c) * 25 + tap] * xv6[c];
                yo[o] += w * XR(o, hy, hx);
            }
        }

        const int cy = py + 2, cx = px + 2;
        float xsC[NSX], uvC[6];
#pragma unroll
        for (int s2 = 0; s2 < NSX; ++s2) xsC[s2] = XR(s2, cy, cx);
#pragma unroll
        for (int c = 0; c < 6; ++c) uvC[c] = XR(NSX + c, cy, cx);

        // svp / Coriolis terms on the vector tendency
        float yv[KL][2];
#pragma unroll
        for (int l = 0; l < KL; ++l) {
            float A0 = 0.0f, A1 = 0.0f;
            for (int s2 = 0; s2 < NSX; ++s2) {
                A0 += svp[(s2 * KL + l) * 2 + 0] * xsC[s2];
                A1 += svp[(s2 * KL + l) * 2 + 1] * xsC[s2];
            }
            float uu = uvC[l * 2 + 0], vv = uvC[l * 2 + 1];
            yv[l][0] = yo[NSX + l * 2 + 0] + A0 * uu + A1 * (-vv);
            yv[l][1] = yo[NSX + l * 2 + 1] + A0 * vv + A1 * uu;
        }

        // vertical physics: omega from div, dT/dp correction
        float divk[KL], dTl[KL], dql[KL], Tk[KL];
#pragma unroll
        for (int l = 0; l < KL; ++l) {
            divk[l] = yo[l * 6 + 3];
            dTl[l]  = yo[l * 6 + 4];
            dql[l]  = yo[l * 6 + 5];
            Tk[l]   = xsC[l * 6 + 4];
        }
        float p0 = plev[0], p1 = plev[1], p2 = plev[2];
        float oh1 = divk[0] * dpv[0];
        float oh2 = oh1 + divk[1] * dpv[1];
        float oh3 = oh2 + divk[2] * dpv[2];
        float om[KL]   = {0.5f * oh1, 0.5f * (oh1 + oh2), 0.5f * (oh2 + oh3)};
        float dTdp[KL] = {(Tk[1] - Tk[0]) / (p1 - p0),
                          (Tk[2] - Tk[0]) / (p2 - p0),
                          (Tk[2] - Tk[1]) / (p2 - p1)};
        const float rcp = RGASC / CPC;
        dTl[0] += om[0] * (rcp * Tk[0] / p0 - dTdp[0]);
        dTl[1] += om[1] * (rcp * Tk[1] / p1 - dTdp[1]);
        dTl[2] += om[2] * (rcp * Tk[2] / p2 - dTdp[2]);

        // RK4 accumulate + next-stage state / final output
        const int gy = ty0 + py, gx = tx0 + px;
        const float c6 = dtf * (1.0f / 6.0f);
        const float ad = aCoef * dtf;
#pragma unroll
        for (int l = 0; l < KL; ++l) {
            size_t iS = ((size_t)(b * KL + l) * HH + gy) * WW + gx;
#pragma unroll
            for (int c = 0; c < 2; ++c) {
                size_t iu = iS * 2 + c;
                float k  = yv[l][c];
                float av = (stage == 0 ? 0.0f : aUV[iu]) + wgt * k;
                aUV[iu] = av;
                if (stage < 3) nUV[iu] = oUV[iu] + ad * k;
                else           out[iu] = oUV[iu] + c6 * av;
            }
            float kT = dTl[l];
            float at_ = (stage == 0 ? 0.0f : aT[iS]) + wgt * kT;
            aT[iS] = at_;
            if (stage < 3) nT[iS] = oT[iS] + ad * kT;
            else           out[OFFT + iS] = oT[iS] + c6 * at_;

            float kq = dql[l];
            float aq_ = (stage == 0 ? 0.0f : aQ[iS]) + wgt * kq;
            aQ[iS] = aq_;
            if (stage < 3) nQ[iS] = oQ[iS] + ad * kq;
            else           out[OFFQ + iS] = oQ[iS] + c6 * aq_;
        }
#undef XR
    }
}

// ---------------------------------------------------------------------------
extern "C" void kernel_launch(void* const* d_in, const int* in_sizes, int n_in,
                              void* d_out, int out_size, void* d_ws, size_t ws_size,
                              hipStream_t stream) {
    (void)in_sizes; (void)n_in; (void)out_size; (void)ws_size;
    const float* uv0  = (const float*)d_in[0];
    const float* T0   = (const float*)d_in[1];
    const float* q0   = (const float*)d_in[2];
    const float* kss  = (const float*)d_in[3];
    const float* kvs  = (const float*)d_in[4];
    const float* ksv  = (const float*)d_in[5];
    const float* kvv  = (const float*)d_in[6];
    const float* kvs2 = (const float*)d_in[7];
    const float* pw   = (const float*)d_in[8];
    const float* svp  = (const float*)d_in[9];
    const float* kew  = (const float*)d_in[10];
    const float* fcor = (const float*)d_in[11];
    const float* plev = (const float*)d_in[12];
    const float* dpv  = (const float*)d_in[13];
    const int*   dtp  = (const int*)d_in[14];

    float* ws = (float*)d_ws;
    size_t off = 0;
    float* Xg  = ws + off; off += (size_t)BB * NCH * PLANE;      // staged field
    float* uvS = ws + off; off += (size_t)BB * KL * PLANE * 2;   // stage state
    float* TS  = ws + off; off += (size_t)BB * KL * PLANE;
    float* qS  = ws + off; off += (size_t)BB * KL * PLANE;
    float* aUV = ws + off; off += (size_t)BB * KL * PLANE * 2;   // RK accumulators
    float* aT  = ws + off; off += (size_t)BB * KL * PLANE;
    float* aQ  = ws + off; off += (size_t)BB * KL * PLANE;
    float* Tm  = ws + off; off += 8;
    float* WA  = ws + off; off += 25 * 7 * 2 * 32 * 2;           // swizzled weights

    wprep_kernel<<<(25 * 7 * 2 * 32 + 255) / 256, 256, 0, stream>>>(kss, kvs, ksv, kvv, pw, WA);

    const float wgt[4]   = {1.0f, 2.0f, 2.0f, 1.0f};
    const float acoef[4] = {0.5f, 0.5f, 1.0f, 0.0f};
    for (int s = 0; s < 4; ++s) {
        const float* su = (s == 0) ? uv0 : uvS;
        const float* sT = (s == 0) ? T0  : TS;
        const float* sq = (s == 0) ? q0  : qS;
        mean_kernel<<<BB * KL, 256, 0, stream>>>(sT, Tm);
        prep_kernel<<<(BB * (int)PLANE + 255) / 256, 256, 0, stream>>>(su, sT, sq, Tm, kew, fcor, Xg);
        dim3 grid(WW / TILEW, HH / TILEH, BB);
        conv_kernel<<<grid, 256, 0, stream>>>(Xg, WA, kvs2, svp, plev, dpv, dtp,
                                              uv0, T0, q0, aUV, aT, aQ, uvS, TS, qS,
                                              (float*)d_out, s, wgt[s], acoef[s]);
    }
}